// FEMGraphNN_60644938219865
// MI455X (gfx1250) — compile-verified
//
#include <hip/hip_runtime.h>
#include <hip/hip_bf16.h>

typedef __attribute__((ext_vector_type(2))) float v2f;
typedef __attribute__((ext_vector_type(8))) float v8f;

#define HID 64

__device__ __forceinline__ void atomAddF(float* p, float v) {
    (void)__hip_atomic_fetch_add(p, v, __ATOMIC_RELAXED, __HIP_MEMORY_SCOPE_AGENT);
}

// ---- degree / normalization -------------------------------------------------

__global__ void k_init_deg(float* __restrict__ deg, int n) {
    int i = blockIdx.x * blockDim.x + threadIdx.x;
    if (i < n) deg[i] = 1.0f;   // self-loop contributes 1 to every node
}

__global__ void k_deg(const int* __restrict__ dst, float* __restrict__ deg, int e) {
    int i = blockIdx.x * blockDim.x + threadIdx.x;
    if (i < e) atomAddF(&deg[dst[i]], 1.0f);
}

__global__ void k_dinv(float* __restrict__ deg, int n) {
    int i = blockIdx.x * blockDim.x + threadIdx.x;
    if (i < n) deg[i] = rsqrtf(deg[i]);   // deg >= 1 always (self-loop)
}

__global__ void k_norm(const int* __restrict__ src, const int* __restrict__ dst,
                       const float* __restrict__ dinv, float* __restrict__ norm, int e) {
    int i = blockIdx.x * blockDim.x + threadIdx.x;
    if (i < e) norm[i] = dinv[src[i]] * dinv[dst[i]];
}

// ---- layer 1 transform: h = x @ W1   (x: [N,2], W1: [2,64]) ----------------

__global__ void k_lin1(const float* __restrict__ x, const float* __restrict__ W1,
                       float* __restrict__ h, int n) {
    int t = blockIdx.x * blockDim.x + threadIdx.x;
    int i = t >> 6, j = t & 63;
    if (i < n) {
        float2 xv = ((const float2*)x)[i];
        h[t] = xv.x * W1[j] + xv.y * W1[HID + j];
    }
}

// ---- propagation: out[i][:] = b[:] + h[i][:]*dinv[i]^2  (bias + self loop) -

__global__ void k_prop_init(const float* __restrict__ h, const float* __restrict__ dinv,
                            const float* __restrict__ b, float* __restrict__ out, int n) {
    int t = blockIdx.x * blockDim.x + threadIdx.x;
    int i = t >> 6, j = t & 63;
    if (i < n) {
        float di = dinv[i];
        out[t] = b[j] + h[t] * di * di;
    }
}

// one wave per edge; each lane handles 2 of the 64 dims via float2
__global__ void k_prop_edges(const int* __restrict__ src, const int* __restrict__ dst,
                             const float* __restrict__ norm, const float* __restrict__ h,
                             float* __restrict__ out, int e) {
    int t  = blockIdx.x * blockDim.x + threadIdx.x;
    int ei = t >> 5;
    if (ei >= e) return;
    int lane = t & 31;
    int s = src[ei], d = dst[ei];
    float w = norm[ei];
    float2 hv = ((const float2*)(h + (size_t)s * HID))[lane];
    float* op = out + (size_t)d * HID + lane * 2;
    atomAddF(op,     hv.x * w);
    atomAddF(op + 1, hv.y * w);
}

// ---- layer 2 GEMM via WMMA: h = relu(in) @ W2  ([N,64] @ [64,64]) ----------
// One wave computes a 16x64 tile: 4 accumulators of 16x16, K looped 4 at a time.
// A frag (16x4 f32): lanes 0-15 hold row M=l, K=k0..k0+1; lanes 16-31 K=k0+2..k0+3.
// B frag (4x16 f32): lanes 0-15 hold N=l for K=k0 (x) / k0+1 (y); lanes 16-31 K=k0+2/k0+3.
// C/D (16x16 f32): VGPR r -> row r (lanes 0-15) / row r+8 (lanes 16-31), col = lane&15.
__global__ void __launch_bounds__(256)
k_gemm_relu_wmma(const float* __restrict__ in, const float* __restrict__ W,
                 float* __restrict__ out, int nrows) {
    int gwave = (blockIdx.x * blockDim.x + threadIdx.x) >> 5;
    int lane  = threadIdx.x & 31;
    int row0  = gwave << 4;
    if (row0 >= nrows) return;   // nrows % 16 == 0, so whole wave branches uniformly
    int hh = lane >> 4;          // half-select: 0 -> K offset 0, 1 -> K offset 2
    int l  = lane & 15;
    v8f acc0 = {}, acc1 = {}, acc2 = {}, acc3 = {};
    const float* arow = in + (size_t)(row0 + l) * HID;
    for (int k0 = 0; k0 < HID; k0 += 4) {
        int ka = k0 + hh * 2;
        v2f a;
        a.x = fmaxf(arow[ka],     0.0f);   // fused relu of previous layer output
        a.y = fmaxf(arow[ka + 1], 0.0f);
        const float* bp = W + (size_t)ka * HID + l;
        v2f b0, b1, b2, b3;
        b0.x = bp[0];  b0.y = bp[HID];
        b1.x = bp[16]; b1.y = bp[HID + 16];
        b2.x = bp[32]; b2.y = bp[HID + 32];
        b3.x = bp[48]; b3.y = bp[HID + 48];
        acc0 = __builtin_amdgcn_wmma_f32_16x16x4_f32(false, a, false, b0, (short)0, acc0, false, false);
        acc1 = __builtin_amdgcn_wmma_f32_16x16x4_f32(false, a, false, b1, (short)0, acc1, false, false);
        acc2 = __builtin_amdgcn_wmma_f32_16x16x4_f32(false, a, false, b2, (short)0, acc2, false, false);
        acc3 = __builtin_amdgcn_wmma_f32_16x16x4_f32(false, a, false, b3, (short)0, acc3, false, false);
    }
    float* orow = out + (size_t)(row0 + hh * 8) * HID + l;
    #pragma unroll
    for (int r = 0; r < 8; ++r) {
        float* o = orow + (size_t)r * HID;
        o[0]  = acc0[r];
        o[16] = acc1[r];
        o[32] = acc2[r];
        o[48] = acc3[r];
    }
}

// ---- layer 3: h3[i] = dot(relu(out[i][:]), W3)  (one wave per node) --------

__global__ void k_lin3_relu(const float* __restrict__ in, const float* __restrict__ W3,
                            float* __restrict__ h3, int n) {
    int gwave = (blockIdx.x * blockDim.x + threadIdx.x) >> 5;
    int lane  = threadIdx.x & 31;
    if (gwave >= n) return;
    const float* row = in + (size_t)gwave * HID;
    float v = fmaxf(row[lane], 0.0f) * W3[lane]
            + fmaxf(row[lane + 32], 0.0f) * W3[lane + 32];
    #pragma unroll
    for (int off = 16; off > 0; off >>= 1) v += __shfl_down(v, off, 32);
    if (lane == 0) h3[gwave] = v;
}

__global__ void k_prop3_init(const float* __restrict__ h3, const float* __restrict__ dinv,
                             const float* __restrict__ b3, float* __restrict__ out, int n) {
    int i = blockIdx.x * blockDim.x + threadIdx.x;
    if (i < n) {
        float di = dinv[i];
        out[i] = b3[0] + h3[i] * di * di;
    }
}

__global__ void k_prop3_edges(const int* __restrict__ src, const int* __restrict__ dst,
                              const float* __restrict__ norm, const float* __restrict__ h3,
                              float* __restrict__ out, int e) {
    int i = blockIdx.x * blockDim.x + threadIdx.x;
    if (i < e) atomAddF(&out[dst[i]], h3[src[i]] * norm[i]);
}

// ---- driver ----------------------------------------------------------------

extern "C" void kernel_launch(void* const* d_in, const int* in_sizes, int n_in,
                              void* d_out, int out_size, void* d_ws, size_t ws_size,
                              hipStream_t stream) {
    const float* x  = (const float*)d_in[0];
    const int*   ei = (const int*)  d_in[1];
    const float* W1 = (const float*)d_in[2];
    const float* b1 = (const float*)d_in[3];
    const float* W2 = (const float*)d_in[4];
    const float* b2 = (const float*)d_in[5];
    const float* W3 = (const float*)d_in[6];
    const float* b3 = (const float*)d_in[7];

    const int N = in_sizes[0] / 2;   // 100000
    const int E = in_sizes[1] / 2;   // 1600000
    const int* src = ei;
    const int* dst = ei + E;

    // workspace layout (all float, offsets stay 8/16-byte aligned)
    float* norm = (float*)d_ws;                 // [E]
    float* dinv = norm + (size_t)E;             // [N]
    float* h    = dinv + (size_t)N;             // [N,64]
    float* out  = h    + (size_t)N * HID;       // [N,64]
    float* h3   = out  + (size_t)N * HID;       // [N]
    float* y    = (float*)d_out;                // [N]

    const int B = 256;
    auto g = [](long long t) { return (int)((t + 255) / 256); };

    // normalization
    k_init_deg<<<g(N), B, 0, stream>>>(dinv, N);
    k_deg     <<<g(E), B, 0, stream>>>(dst, dinv, E);
    k_dinv    <<<g(N), B, 0, stream>>>(dinv, N);
    k_norm    <<<g(E), B, 0, stream>>>(src, dst, dinv, norm, E);

    // layer 1: transform + propagate
    k_lin1      <<<g((long long)N * HID), B, 0, stream>>>(x, W1, h, N);
    k_prop_init <<<g((long long)N * HID), B, 0, stream>>>(h, dinv, b1, out, N);
    k_prop_edges<<<g((long long)E * 32),  B, 0, stream>>>(src, dst, norm, h, out, E);

    // layer 2: relu + WMMA GEMM + propagate
    k_gemm_relu_wmma<<<g((long long)(N / 16) * 32), 256, 0, stream>>>(out, W2, h, N);
    k_prop_init <<<g((long long)N * HID), B, 0, stream>>>(h, dinv, b2, out, N);
    k_prop_edges<<<g((long long)E * 32),  B, 0, stream>>>(src, dst, norm, h, out, E);

    // layer 3: relu + dot with W3 + propagate (scalar width)
    k_lin3_relu  <<<g((long long)N * 32), B, 0, stream>>>(out, W3, h3, N);
    k_prop3_init <<<g(N), B, 0, stream>>>(h3, dinv, b3, y, N);
    k_prop3_edges<<<g(E), B, 0, stream>>>(src, dst, norm, h3, y, E);
}